// _proposal_47141561040897
// MI455X (gfx1250) — compile-verified
//
#include <hip/hip_runtime.h>
#include <hip/hip_bf16.h>
#include <stdint.h>

// Problem constants (match reference setup_inputs)
constexpr int B    = 8;
constexpr int N    = 16384;      // boxes per batch
constexpr int NW   = N / 32;     // 512 keep-mask words per batch
constexpr int POST = 1000;
constexpr float NMS_T = 0.7f;

typedef unsigned int u32x4 __attribute__((ext_vector_type(4)));
typedef int          i32x4 __attribute__((ext_vector_type(4)));
typedef int          i32x8 __attribute__((ext_vector_type(8)));

// ---------------------------------------------------------------------------
// CDNA5 helpers: async global->LDS, TDM tensor DMA, counter waits, ballot
// ---------------------------------------------------------------------------
__device__ __forceinline__ void async_copy_b128_to_lds(void* lds_dst, const void* gsrc) {
    // GLOBAL_LOAD_ASYNC_TO_LDS_B128: VDST = LDS byte address (VGPR),
    // VADDR = 64-bit global address (VGPR pair). Tracked with ASYNCcnt.
    unsigned           lds = (unsigned)(uintptr_t)lds_dst;      // addrspace(3) offset
    unsigned long long ga  = (unsigned long long)(uintptr_t)gsrc;
    asm volatile("global_load_async_to_lds_b128 %0, %1, off"
                 :
                 : "v"(lds), "v"(ga)
                 : "memory");
}

__device__ __forceinline__ void wait_async0() {
#if __has_builtin(__builtin_amdgcn_s_wait_asynccnt)
    __builtin_amdgcn_s_wait_asynccnt(0);
#else
    asm volatile("s_wait_asynccnt 0" ::: "memory");
#endif
}

__device__ __forceinline__ void wait_tensor0() {
#if __has_builtin(__builtin_amdgcn_s_wait_tensorcnt)
    __builtin_amdgcn_s_wait_tensorcnt(0);
#else
    asm volatile("s_wait_tensorcnt 0" ::: "memory");
#endif
}

__device__ __forceinline__ unsigned ballot32(bool p) {
#if __has_builtin(__builtin_amdgcn_ballot_w32)
    return __builtin_amdgcn_ballot_w32(p);
#else
    return (unsigned)__ballot(p);
#endif
}

#if __has_builtin(__builtin_amdgcn_tensor_load_to_lds)
#define HAVE_TDM 1
// One-shot TDM DMA of a contiguous `bytes` region (multiple of 8) into LDS.
// D# per CDNA5 ISA ch.8: 1D tile of 8-byte elements, tile_dim0 = bytes/8.
__device__ __forceinline__ void tdm_load_contig_to_lds(void* lds_dst, const void* gsrc,
                                                       unsigned bytes) {
    const unsigned           lds = (unsigned)(uintptr_t)lds_dst;
    const unsigned long long ga  = (unsigned long long)(uintptr_t)gsrc;
    const unsigned           n8  = bytes >> 3;          // 8-byte elements (<= 65535)

    u32x4 g0;
    g0.x = 1u;                                          // count=1 (valid), user mode
    g0.y = lds;                                         // lds_addr (bytes)
    g0.z = (unsigned)(ga & 0xFFFFFFFFull);              // global_addr[31:0]
    g0.w = (unsigned)((ga >> 32) & 0x01FFFFFFull)       // global_addr[56:32]
         | (2u << 30);                                  // type = 2 ("image")

    i32x8 g1;
    g1[0] = (int)(3u << 16);                            // wg_mask=0, data_size=3 (8B)
    g1[1] = (int)((n8 & 0xFFFFu) << 16);                // tensor_dim0[15:0]
    g1[2] = (int)((n8 >> 16) | (1u << 16));             // tensor_dim0[31:16] | tensor_dim1=1
    g1[3] = (int)((n8 & 0xFFFFu) << 16);                // tile_dim0 = n8
    g1[4] = 1;                                          // tile_dim1 = 1
    g1[5] = (int)n8;                                    // tensor_dim0_stride[31:0]
    g1[6] = 0;                                          // stride hi / dim1_stride lo
    g1[7] = 0;

    i32x4 g2 = (i32x4)0;                                // <=2D tensor: groups 2/3 unused
    i32x4 g3 = (i32x4)0;
#if __clang_major__ >= 23
    i32x8 g4 = (i32x8)0;
    __builtin_amdgcn_tensor_load_to_lds(g0, g1, g2, g3, g4, 0);
#else
    __builtin_amdgcn_tensor_load_to_lds(g0, g1, g2, g3, 0);
#endif
}
#else
#define HAVE_TDM 0
#endif

// ---------------------------------------------------------------------------
// Kernel 1: decode boxes + build sortable 64-bit keys
//   key = (sortable(score) << 32) | (0xFFFFFFFF - n)
//   descending sort on key => descending score, ties broken by ascending n
// ---------------------------------------------------------------------------
__global__ void decode_kernel(const float* __restrict__ fg,
                              const float* __restrict__ reg,
                              const float* __restrict__ anc,
                              const int* __restrict__ imgH,
                              const int* __restrict__ imgW,
                              float4* __restrict__ boxes,
                              unsigned long long* __restrict__ keys) {
    int gid = blockIdx.x * blockDim.x + threadIdx.x;
    if (gid >= B * N) return;

    const float Wm1 = (float)(*imgW - 1);
    const float Hm1 = (float)(*imgH - 1);

    float4 a = ((const float4*)anc)[gid];   // sx, sy, sw, sh
    float4 o = ((const float4*)reg)[gid];   // ox, oy, ow, oh

    float cx = fmaf(a.z, o.x, a.x);
    float cy = fmaf(a.w, o.y, a.y);
    float w  = a.z * __expf(o.z);
    float h  = a.w * __expf(o.w);

    float x1 = fminf(fmaxf(cx - 0.5f * w, 0.0f), Wm1);
    float y1 = fminf(fmaxf(cy - 0.5f * h, 0.0f), Hm1);
    float x2 = fminf(fmaxf(cx + 0.5f * w, 0.0f), Wm1);
    float y2 = fminf(fmaxf(cy + 0.5f * h, 0.0f), Hm1);

    boxes[gid] = make_float4(x1, y1, x2, y2);

    unsigned u = __float_as_uint(fg[gid]);
    u ^= (u >> 31) ? 0xFFFFFFFFu : 0x80000000u;          // order-preserving map
    unsigned n = (unsigned)gid & (unsigned)(N - 1);
    keys[gid]  = ((unsigned long long)u << 32) |
                 (unsigned long long)(0xFFFFFFFFu - n);   // stable tie-break
}

// ---------------------------------------------------------------------------
// Kernel 2: per-batch bitonic sort, fully in LDS (16384 x u64 = 128 KB).
// One 1024-thread workgroup per batch; keys staged via async B128 DMA.
// Epilogue gathers boxes into sorted order.
// ---------------------------------------------------------------------------
__global__ void __launch_bounds__(1024)
sort_kernel(const unsigned long long* __restrict__ keys,
            const float4* __restrict__ boxes,
            float4* __restrict__ boxesSorted) {
    extern __shared__ char dynsm[];
    unsigned long long* lkey = (unsigned long long*)dynsm;

    const int b   = blockIdx.x;
    const int tid = threadIdx.x;
    const unsigned long long* kb = keys + (size_t)b * N;

    // Async stage: two u64 keys per B128 transfer, 8 per thread.
    for (int t = tid; t < N / 2; t += 1024)
        async_copy_b128_to_lds(&lkey[2 * t], &kb[2 * t]);
    wait_async0();
    __syncthreads();

    // Bitonic sort, descending. 105 (k,j) passes, 8 compare-exchanges/thread.
    for (unsigned k = 2; k <= (unsigned)N; k <<= 1) {
        for (unsigned j = k >> 1; j > 0; j >>= 1) {
            for (unsigned t = tid; t < (unsigned)N / 2; t += 1024) {
                unsigned i = ((t & ~(j - 1)) << 1) | (t & (j - 1));
                unsigned p = i + j;
                unsigned long long x = lkey[i];
                unsigned long long y = lkey[p];
                bool desc = ((i & k) == 0);
                bool sw   = desc ? (x < y) : (x > y);
                if (sw) { lkey[i] = y; lkey[p] = x; }
            }
            __syncthreads();
        }
    }

    // Gather boxes into sorted order.
    for (int t = tid; t < N; t += 1024) {
        unsigned orig = 0xFFFFFFFFu - (unsigned)lkey[t];
        boxesSorted[(size_t)b * N + t] = boxes[(size_t)b * N + orig];
    }
}

// ---------------------------------------------------------------------------
// Kernel 3: greedy NMS. One workgroup per batch; ALL 16384 boxes live in LDS
// (256 KB, CDNA5-only). The whole tile is staged by ONE Tensor-Data-Mover DMA
// (tensor_load_to_lds, TENSORcnt) issued by wave 0 while the other waves set
// up the keep bitmap. Suppression: wave32 ballot + single ds_and_b32 per word.
// Epilogue: popcount scan -> compact top-1000 into d_out.
// ---------------------------------------------------------------------------
__global__ void __launch_bounds__(1024)
nms_kernel(const float4* __restrict__ boxesSorted,
           float* __restrict__ outBoxes,          // (B, POST, 4)
           int* __restrict__ outIdx,              // (B, POST) rank in sorted order
           unsigned char* __restrict__ outValid)  // (B, POST)
{
    extern __shared__ char dynsm[];
    float4* lbox = (float4*)dynsm;                 // 16384 * 16 B = 256 KB

    __shared__ unsigned keepw[NW];                 // 2 KB
    __shared__ unsigned offs[NW + 1];

    const int b    = blockIdx.x;
    const int tid  = threadIdx.x;
    const int lane = tid & 31;
    const int warp = tid >> 5;                     // 32 wave32s

    const float4* src = boxesSorted + (size_t)b * N;

#if HAVE_TDM
    if (tid < 32) {                                // wave 0 launches one TDM DMA
        tdm_load_contig_to_lds(lbox, src, (unsigned)N * 16u);
    }
    if (tid < NW) keepw[tid] = 0xFFFFFFFFu;        // overlap with DMA
    if (tid < 32) wait_tensor0();
#else
    for (int t = tid; t < N; t += 1024)
        async_copy_b128_to_lds(&lbox[t], &src[t]);
    if (tid < NW) keepw[tid] = 0xFFFFFFFFu;
    wait_async0();
#endif
    __syncthreads();

    // Greedy serial pass over sorted boxes.
    for (int i = 0; i < N - 1; ++i) {
        if ((keepw[i >> 5] >> (i & 31)) & 1u) {    // uniform branch (LDS bcast)
            float4 bi = lbox[i];
            float  ai = (bi.z - bi.x) * (bi.w - bi.y);
            for (int w = (i >> 5) + warp; w < NW; w += 32) {
                unsigned kw  = keepw[w];
                int      j   = (w << 5) | lane;
                bool     sup = false;
                if (j > i && ((kw >> lane) & 1u)) {
                    float4 bj  = lbox[j];
                    float xx1  = fmaxf(bi.x, bj.x);
                    float yy1  = fmaxf(bi.y, bj.y);
                    float xx2  = fminf(bi.z, bj.z);
                    float yy2  = fminf(bi.w, bj.w);
                    float iw   = fmaxf(xx2 - xx1, 0.0f);
                    float ih   = fmaxf(yy2 - yy1, 0.0f);
                    float inter = iw * ih;
                    float aj   = (bj.z - bj.x) * (bj.w - bj.y);
                    // iou > T  <=>  inter > T * union (union > 0), avoids divide
                    sup = inter > NMS_T * (ai + aj - inter + 1e-9f);
                }
                unsigned m = ballot32(sup);
                if (m && lane == 0) atomicAnd(&keepw[w], ~m);
            }
            __syncthreads();
        }
    }
    __syncthreads();

    // Exclusive scan of per-word popcounts (512 trivial adds, thread 0).
    if (tid == 0) {
        unsigned acc = 0;
        for (int w = 0; w < NW; ++w) { offs[w] = acc; acc += __popc(keepw[w]); }
        offs[NW] = acc;
    }
    __syncthreads();

    const unsigned totalc = offs[NW] < (unsigned)POST ? offs[NW] : (unsigned)POST;

    // Emit kept entries (ranks [0, totalc)); one word per thread.
    if (tid < NW) {
        unsigned kw   = keepw[tid];
        unsigned rank = offs[tid];
        while (kw && rank < (unsigned)POST) {
            int bit = __ffs(kw) - 1;
            kw &= kw - 1;
            int j = (tid << 5) | bit;
            ((float4*)outBoxes)[(size_t)b * POST + rank] = lbox[j];
            outIdx[b * POST + rank]   = j;    // position in score-sorted order
            outValid[b * POST + rank] = 1;
            ++rank;
        }
    }
    // Fill invalid tail (ranks [totalc, POST)) — disjoint from kept writes.
    for (int r = tid; r < POST; r += 1024) {
        if ((unsigned)r >= totalc) {
            ((float4*)outBoxes)[(size_t)b * POST + r] = make_float4(0.f, 0.f, 0.f, 0.f);
            outIdx[b * POST + r]   = -1;
            outValid[b * POST + r] = 0;
        }
    }
}

// ---------------------------------------------------------------------------
// Launcher
// ---------------------------------------------------------------------------
extern "C" void kernel_launch(void* const* d_in, const int* in_sizes, int n_in,
                              void* d_out, int out_size, void* d_ws, size_t ws_size,
                              hipStream_t stream) {
    const float* fg   = (const float*)d_in[0];   // (B, N)
    const float* reg  = (const float*)d_in[1];   // (B, N, 4)
    const float* anc  = (const float*)d_in[2];   // (B, N, 4)
    const int*   imgH = (const int*)d_in[3];
    const int*   imgW = (const int*)d_in[4];

    char* ws = (char*)d_ws;                                   // usage: ~5 MB
    float4*             boxes       = (float4*)ws;                               // 2 MB
    float4*             boxesSorted = (float4*)(ws + (size_t)B * N * 16);        // 2 MB
    unsigned long long* keys        = (unsigned long long*)(ws + (size_t)B * N * 32); // 1 MB

    // Outputs concatenated flat in return order: boxes_out f32, kept_idx i32, valid bool
    float*         outBoxes = (float*)d_out;
    int*           outIdx   = (int*)((char*)d_out + (size_t)B * POST * 4 * sizeof(float));
    unsigned char* outValid = (unsigned char*)((char*)outIdx + (size_t)B * POST * sizeof(int));

    decode_kernel<<<(B * N) / 256, 256, 0, stream>>>(fg, reg, anc, imgH, imgW, boxes, keys);
    sort_kernel<<<B, 1024, (size_t)N * sizeof(unsigned long long), stream>>>(keys, boxes, boxesSorted);
    nms_kernel<<<B, 1024, (size_t)N * sizeof(float4), stream>>>(boxesSorted, outBoxes, outIdx, outValid);
}